// MSTAGNN_7069516169750
// MI455X (gfx1250) — compile-verified
//
#include <hip/hip_runtime.h>
#include <hip/hip_bf16.h>

// ---------------------------------------------------------------------------
// MSTAGNN for MI455X (gfx1250, wave32).
// Strategy: CSR-by-destination counting sort -> all hops are gather (no float
// atomics, L2-resident working set). QKV projections via V_WMMA_F32_16X16X4_F32.
// ---------------------------------------------------------------------------

#define NNODES 10000
#define NEDGES 160000
#define HID    64
#define NHEADS 8
#define HDIM   8

typedef __attribute__((ext_vector_type(2))) float v2f;
typedef __attribute__((ext_vector_type(8))) float v8f;

// ---------------------------------------------------------------- utilities
__global__ void zero_ints_kernel(int* p, int n) {
    int i = blockIdx.x * blockDim.x + threadIdx.x;
    for (; i < n; i += gridDim.x * blockDim.x) p[i] = 0;
}

// deg[c] += 1 for each edge destination c
__global__ void degree_kernel(const int* __restrict__ col, int* __restrict__ deg, int e) {
    int i = blockIdx.x * blockDim.x + threadIdx.x;
    for (; i < e; i += gridDim.x * blockDim.x) atomicAdd(&deg[col[i]], 1);
}

// single-block exclusive scan of deg -> offsets[0..n], offsets[n] = total
__global__ __launch_bounds__(1024) void scan_kernel(const int* __restrict__ deg,
                                                    int* __restrict__ offsets, int n) {
    __shared__ int partial[1024];
    int t = threadIdx.x;
    int chunk = (n + 1023) / 1024;
    int beg = t * chunk;
    int end = beg + chunk < n ? beg + chunk : n;
    int s = 0;
    for (int i = beg; i < end; ++i) s += deg[i];
    partial[t] = s;
    __syncthreads();
    for (int off = 1; off < 1024; off <<= 1) {
        int v = (t >= off) ? partial[t - off] : 0;
        __syncthreads();
        partial[t] += v;
        __syncthreads();
    }
    int run = (t == 0) ? 0 : partial[t - 1];
    for (int i = beg; i < end; ++i) { offsets[i] = run; run += deg[i]; }
    if (t == 0) offsets[n] = partial[1023];
}

// bucket fill: csr holds edge ids grouped by destination node
__global__ void fill_csr_kernel(const int* __restrict__ col, const int* __restrict__ offsets,
                                int* __restrict__ cursor, int* __restrict__ csr, int e) {
    int i = blockIdx.x * blockDim.x + threadIdx.x;
    for (; i < e; i += gridDim.x * blockDim.x) {
        int c = col[i];
        int p = atomicAdd(&cursor[c], 1);
        csr[offsets[c] + p] = i;
    }
}

// norm[e] = deg^-0.5[row] * deg^-0.5[col] * edge_weight[e]
__global__ void norm_kernel(const int* __restrict__ row, const int* __restrict__ col,
                            const int* __restrict__ deg, const float* __restrict__ ew,
                            float* __restrict__ nrm, int e) {
    int i = blockIdx.x * blockDim.x + threadIdx.x;
    for (; i < e; i += gridDim.x * blockDim.x) {
        int dr = deg[row[i]], dc = deg[col[i]];
        float ir = dr > 0 ? 1.0f / sqrtf((float)dr) : 0.0f;
        float ic = dc > 0 ? 1.0f / sqrtf((float)dc) : 0.0f;
        nrm[i] = ir * ic * ew[i];
    }
}

// ------------------------------------------------- QKV projection with WMMA
// grid = (N/16, 3); block = 128 (4 waves). Wave w computes 16x16 tile at
// column base w*16 of matrix blockIdx.y (0:Q relu, 1:K, 2:V).
// V_WMMA_F32_16X16X4_F32: A 16x4 (2 VGPR), B 4x16 (2 VGPR), C/D 16x16 (8 VGPR).
__global__ __launch_bounds__(128) void qkv_wmma_kernel(
    const float* __restrict__ x,
    const float* __restrict__ Wq, const float* __restrict__ Wk, const float* __restrict__ Wv,
    const float* __restrict__ bq, const float* __restrict__ bk, const float* __restrict__ bv,
    float* __restrict__ Q, float* __restrict__ K, float* __restrict__ V) {
    const int wave = threadIdx.x >> 5;
    const int lane = threadIdx.x & 31;
    const int rowBase = blockIdx.x * 16;
    const int colBase = wave * 16;
    const int which = blockIdx.y;

    const float* W = which == 0 ? Wq : (which == 1 ? Wk : Wv);
    const float* bias = which == 0 ? bq : (which == 1 ? bk : bv);
    float* out = which == 0 ? Q : (which == 1 ? K : V);

    const int m = lane & 15;            // output row within tile (A rows)
    const int n = lane & 15;            // output col within tile (B cols)
    const int koff = (lane >> 4) * 2;   // lanes 16-31 carry K+2/K+3 halves

    v8f acc = {};
    const float* xrow = x + (size_t)(rowBase + m) * HID;
#pragma unroll
    for (int kk = 0; kk < HID; kk += 4) {
        // A fragment: 16x4 f32; VGPR0 = K {0|2}, VGPR1 = K {1|3}
        v2f a;
        float2 ax = *(const float2*)(xrow + kk + koff);
        a.x = ax.x; a.y = ax.y;
        // B fragment: 4x16 f32 (row-major W[k][n]); mirrored K striping
        v2f b;
        b.x = W[(size_t)(kk + koff) * HID + colBase + n];
        b.y = W[(size_t)(kk + koff + 1) * HID + colBase + n];
        acc = __builtin_amdgcn_wmma_f32_16x16x4_f32(
            false, a, false, b, (short)0, acc, false, false);
    }

    const float bn = bias[colBase + n];
    const int mhalf = (lane >> 4) * 8;  // C/D: lanes 16-31 hold M=r+8
#pragma unroll
    for (int r = 0; r < 8; ++r) {
        float v = acc[r] + bn;
        if (which == 0) v = fmaxf(v, 0.0f);   // relu for Q only
        out[(size_t)(rowBase + r + mhalf) * HID + colBase + n] = v;
    }
}

// ------------------------------------------------------------- hop 0 gather
// One block of 64 threads per destination node c; thread t = h*8+i.
// M[c,h,i,j] = sum_{e in in(c)} norm[e] * relu(K[row]+ef[e])[h,i] * V[row][h,j]
__global__ __launch_bounds__(64) void hop0_kernel(
    const float* __restrict__ K, const float* __restrict__ V,
    const float* __restrict__ ef, const float* __restrict__ nrm,
    const int* __restrict__ row, const int* __restrict__ offsets,
    const int* __restrict__ csr, float* __restrict__ M) {
    const int c = blockIdx.x;
    const int t = threadIdx.x;          // t = h*8 + i
    const int h = t >> 3;
    float acc[8] = {0, 0, 0, 0, 0, 0, 0, 0};

    const int beg = offsets[c];
    const int end = offsets[c + 1];
    for (int p = beg; p < end; ++p) {
        const int e = csr[p];
        const int r = row[e];
        if (p + 1 < end) {
            int r2 = row[csr[p + 1]];
            __builtin_prefetch(K + (size_t)r2 * HID + t, 0, 3);
            __builtin_prefetch(V + (size_t)r2 * HID + t, 0, 3);
        }
        const float w = nrm[e];
        float kj = K[(size_t)r * HID + t] + ef[(size_t)e * HID + t];
        kj = fmaxf(kj, 0.0f);
        const float wk = w * kj;
        const float4* vr = (const float4*)(V + (size_t)r * HID + h * HDIM);
        float4 va = vr[0], vb = vr[1];
        acc[0] = fmaf(wk, va.x, acc[0]);
        acc[1] = fmaf(wk, va.y, acc[1]);
        acc[2] = fmaf(wk, va.z, acc[2]);
        acc[3] = fmaf(wk, va.w, acc[3]);
        acc[4] = fmaf(wk, vb.x, acc[4]);
        acc[5] = fmaf(wk, vb.y, acc[5]);
        acc[6] = fmaf(wk, vb.z, acc[6]);
        acc[7] = fmaf(wk, vb.w, acc[7]);
    }
    float* dst = M + (size_t)c * 512 + (size_t)t * 8;
#pragma unroll
    for (int j = 0; j < 8; ++j) dst[j] = acc[j];
}

// ------------------------------------------------- hops 1..2: M propagation
// Mout[c] = sum_{e in in(c)} norm[e] * Min[row[e]]   (512 floats per node)
// One block of 256 threads per node, float2 per thread.
__global__ __launch_bounds__(256) void propagate_kernel(
    const float* __restrict__ Min, float* __restrict__ Mout,
    const float* __restrict__ nrm, const int* __restrict__ row,
    const int* __restrict__ offsets, const int* __restrict__ csr) {
    const int c = blockIdx.x;
    const int t = threadIdx.x;
    float a0 = 0.0f, a1 = 0.0f;
    const int beg = offsets[c];
    const int end = offsets[c + 1];
    for (int p = beg; p < end; ++p) {
        const int e = csr[p];
        const int r = row[e];
        if (p + 1 < end) {
            int r2 = row[csr[p + 1]];
            __builtin_prefetch(Min + (size_t)r2 * 512 + 2 * t, 0, 3);
        }
        const float w = nrm[e];
        float2 v = *(const float2*)(Min + (size_t)r * 512 + 2 * t);
        a0 = fmaf(w, v.x, a0);
        a1 = fmaf(w, v.y, a1);
    }
    float2* dst = (float2*)(Mout + (size_t)c * 512 + 2 * t);
    *dst = make_float2(a0, a1);
}

// ------------------------------- per-hop output: H = Q*M, clip-norm, accum
// One block of 64 threads per node; thread t = h*8 + j.
__global__ __launch_bounds__(64) void apply_hop_kernel(
    const float* __restrict__ Q, const float* __restrict__ M,
    const float* __restrict__ V, const float* __restrict__ hopwise,
    int hopidx, int init, float* __restrict__ out) {
    const int c = blockIdx.x;
    const int t = threadIdx.x;
    const int h = t >> 3;
    const int j = t & 7;

    const float* m = M + (size_t)c * 512 + (size_t)h * 64;  // [i][j]
    const float* q = Q + (size_t)c * HID + h * HDIM;
    float H = 0.0f;
#pragma unroll
    for (int i = 0; i < HDIM; ++i) H = fmaf(q[i], m[i * 8 + j], H);

    // nrm = sqrt(sum_j (H/sqrt(8))^2) reduced over the 8 lanes of this head
    float s = H * H;
    s += __shfl_xor(s, 1, 32);
    s += __shfl_xor(s, 2, 32);
    s += __shfl_xor(s, 4, 32);
    float nn = sqrtf(s * 0.125f);
    if (nn > 1.0f) H /= nn;

    const float wk = hopwise[hopidx];
    const size_t oi = (size_t)c * HID + t;
    float base = init ? hopwise[0] * V[oi] : out[oi];
    out[oi] = fmaf(wk, H, base);
}

// ---------------------------------------------------------------------------
extern "C" void kernel_launch(void* const* d_in, const int* in_sizes, int n_in,
                              void* d_out, int out_size, void* d_ws, size_t ws_size,
                              hipStream_t stream) {
    const float* x  = (const float*)d_in[0];
    const int* ei   = (const int*)d_in[1];
    const float* ef = (const float*)d_in[2];
    const float* ew = (const float*)d_in[3];
    const float* Wq = (const float*)d_in[4];
    const float* bq = (const float*)d_in[5];
    const float* Wk = (const float*)d_in[6];
    const float* bk = (const float*)d_in[7];
    const float* Wv = (const float*)d_in[8];
    const float* bv = (const float*)d_in[9];
    const float* hopwise = (const float*)d_in[10];

    const int N = NNODES, E = NEDGES;
    const int* row = ei;
    const int* col = ei + E;
    float* out = (float*)d_out;

    // ---- workspace carve-up (256B aligned regions) ----
    size_t off = 0;
    auto carve = [&](size_t bytes) {
        size_t o = off;
        off += (bytes + 255) & ~(size_t)255;
        return (char*)d_ws + o;
    };
    int*   deg     = (int*)carve((size_t)N * 4);
    int*   cursor  = (int*)carve((size_t)N * 4);
    int*   offsets = (int*)carve((size_t)(N + 1) * 4);
    int*   csr     = (int*)carve((size_t)E * 4);
    float* nrm     = (float*)carve((size_t)E * 4);
    float* Q       = (float*)carve((size_t)N * HID * 4);
    float* K       = (float*)carve((size_t)N * HID * 4);
    float* V       = (float*)carve((size_t)N * HID * 4);
    float* M0      = (float*)carve((size_t)N * 512 * 4);
    float* M1      = (float*)carve((size_t)N * 512 * 4);
    (void)ws_size; (void)in_sizes; (void)n_in; (void)out_size;

    const int tpb = 256;
    const int eblk = (E + tpb - 1) / tpb;
    const int nblk = (N + tpb - 1) / tpb;

    // 1) CSR-by-destination counting sort + symmetric-norm edge weights
    zero_ints_kernel<<<nblk, tpb, 0, stream>>>(deg, N);
    zero_ints_kernel<<<nblk, tpb, 0, stream>>>(cursor, N);
    degree_kernel<<<eblk, tpb, 0, stream>>>(col, deg, E);
    scan_kernel<<<1, 1024, 0, stream>>>(deg, offsets, N);
    fill_csr_kernel<<<eblk, tpb, 0, stream>>>(col, offsets, cursor, csr, E);
    norm_kernel<<<eblk, tpb, 0, stream>>>(row, col, deg, ew, nrm, E);

    // 2) QKV projections: V_WMMA_F32_16X16X4_F32, one wave per 16x16 tile
    qkv_wmma_kernel<<<dim3(N / 16, 3), 128, 0, stream>>>(
        x, Wq, Wk, Wv, bq, bk, bv, Q, K, V);

    // 3) hop 0: gather outer-product messages into M0, then H + hidden init
    hop0_kernel<<<N, 64, 0, stream>>>(K, V, ef, nrm, row, offsets, csr, M0);
    apply_hop_kernel<<<N, 64, 0, stream>>>(Q, M0, V, hopwise, 1, 1, out);

    // 4) hops 1..2: propagate M, apply H each hop
    propagate_kernel<<<N, 256, 0, stream>>>(M0, M1, nrm, row, offsets, csr);
    apply_hop_kernel<<<N, 64, 0, stream>>>(Q, M1, V, hopwise, 2, 0, out);

    propagate_kernel<<<N, 256, 0, stream>>>(M1, M0, nrm, row, offsets, csr);
    apply_hop_kernel<<<N, 64, 0, stream>>>(Q, M0, V, hopwise, 3, 0, out);
}